// GE2ELossSup_11527692223242
// MI455X (gfx1250) — compile-verified
//
#include <hip/hip_runtime.h>
#include <math.h>

// Problem constants from the reference
#define N_SPK 1024
#define M_UTT 32
#define D_DIM 256
#define R_ROWS (N_SPK * M_UTT)   // 32768
#define EPS_C 1e-6f

typedef float v2f __attribute__((ext_vector_type(2)));
typedef float v8f __attribute__((ext_vector_type(8)));
typedef unsigned int u32x4 __attribute__((ext_vector_type(4)));
typedef int i32x4 __attribute__((ext_vector_type(4)));
typedef int i32x8 __attribute__((ext_vector_type(8)));

// ---------------------------------------------------------------------------
// Kernel 1: per-speaker statistics.
//   cent[n][d] = mean_m dvecs[n][m][d]
//   rcn[n]     = 1/||cent[n]||
//   rdn[r]     = 1/||dvecs[r]||            (r = n*M + m)
//   tlog[r]    = w*clip(cos_loo, eps) + b  (the diagonal/target logit)
// cos_loo = (x.S - ||x||^2) / (||x|| * ||S - x||)   [(M-1) factors cancel]
// ---------------------------------------------------------------------------
__global__ __launch_bounds__(256) void k_stats(
    const float* __restrict__ dvecs,
    const float* __restrict__ wp, const float* __restrict__ bp,
    float* __restrict__ cent, float* __restrict__ rcn,
    float* __restrict__ rdn, float* __restrict__ tlog)
{
    __shared__ float xs[M_UTT][D_DIM];   // 32 KB
    __shared__ float Ss[D_DIM];
    __shared__ float red[8];
    __shared__ float snormsq_s;

    const int n = blockIdx.x;
    const int t = threadIdx.x;
    const int wave = t >> 5;
    const int lane = t & 31;

    {
        const float4* s4 = (const float4*)(dvecs + (size_t)n * M_UTT * D_DIM);
        float4* d4 = (float4*)&xs[0][0];
#pragma unroll
        for (int i = 0; i < 8; ++i) d4[t + 256 * i] = s4[t + 256 * i];
    }
    __syncthreads();

    const int d = t;
    float s = 0.f;
#pragma unroll
    for (int m = 0; m < M_UTT; ++m) s += xs[m][d];
    Ss[d] = s;
    cent[(size_t)n * D_DIM + d] = s * (1.0f / M_UTT);

    float p = s * s;
#pragma unroll
    for (int off = 16; off > 0; off >>= 1) p += __shfl_xor(p, off, 32);
    if (lane == 0) red[wave] = p;
    __syncthreads();
    if (t == 0) {
        float tot = 0.f;
#pragma unroll
        for (int i = 0; i < 8; ++i) tot += red[i];
        snormsq_s = tot;
        rcn[n] = (float)M_UTT * rsqrtf(tot);   // 1/||S/M|| = M/||S||
    }
    __syncthreads();

    const float snormsq = snormsq_s;
    const float w = *wp, b = *bp;

#pragma unroll
    for (int mm = 0; mm < 4; ++mm) {
        const int m = wave + mm * 8;
        float dot = 0.f, nsq = 0.f;
#pragma unroll
        for (int i = 0; i < 8; ++i) {
            const float x = xs[m][lane + 32 * i];
            dot = fmaf(x, Ss[lane + 32 * i], dot);
            nsq = fmaf(x, x, nsq);
        }
#pragma unroll
        for (int off = 16; off > 0; off >>= 1) {
            dot += __shfl_xor(dot, off, 32);
            nsq += __shfl_xor(nsq, off, 32);
        }
        if (lane == 0) {
            const int r = n * M_UTT + m;
            const float dn = sqrtf(nsq);
            rdn[r] = 1.0f / dn;
            const float loo2 = snormsq - 2.0f * dot + nsq;   // ||S-x||^2
            float cl = (dot - nsq) / (dn * sqrtf(loo2));
            cl = fmaxf(cl, EPS_C);
            tlog[r] = fmaf(w, cl, b);
        }
    }
}

// ---------------------------------------------------------------------------
// TDM: DMA one 64x256 f32 tile of centroids from global into LDS, inserting
// a 4-DWORD pad after every 256 DWORDs (row) -> rows land at stride 260
// floats, matching the bank-conflict-free lB layout.
// D# per CDNA5 ISA:
//  g0: count=1 | lds_addr | global_addr[56:0] | type=2
//  g1: data_size=4B, pad_enable, pad_interval=256dw(code 7), pad_amount=4dw
//      (code 3), tensor_dim0=256, tensor_dim1=1024, tile_dim0=256,
//      tile_dim1=64, tensor_dim0_stride=256
//  g2/g3: zero (2-D tensor)
// ---------------------------------------------------------------------------
__device__ static inline void tdm_load_b_tile(unsigned lds_byte_off,
                                              const float* gsrc)
{
    const unsigned long long ga = (unsigned long long)(uintptr_t)gsrc;
    u32x4 g0;
    g0[0] = 1u;                                        // count=1 (user D#)
    g0[1] = lds_byte_off;                              // LDS byte address
    g0[2] = (unsigned)ga;                              // global_addr[31:0]
    g0[3] = (unsigned)((ga >> 32) & 0x01FFFFFFu)       // global_addr[56:32]
            | (2u << 30);                              // type=2 ("image")
    i32x8 g1;
    g1[0] = (int)((2u << 16)      // data_size: 4 bytes
                  | (1u << 20)    // pad_enable
                  | (7u << 22)    // pad_interval: 256 DWORDs
                  | (3u << 25));  // pad_amount: 4 DWORDs
    g1[1] = (int)(256u << 16);    // tensor_dim0 = 256 (bits 79:48, low half)
    g1[2] = (int)(1024u << 16);   // tensor_dim0 hi=0 | tensor_dim1 = 1024
    g1[3] = (int)(256u << 16);    // tensor_dim1 hi=0 | tile_dim0 = 256
    g1[4] = 64;                   // tile_dim1 = 64 | tile_dim2 = 0
    g1[5] = 256;                  // tensor_dim0_stride[31:0] = 256 elems
    g1[6] = 0;                    // stride hi | tensor_dim1_stride lo
    g1[7] = 0;                    // tensor_dim1_stride hi
    i32x4 g2 = {0, 0, 0, 0};
    i32x4 g3 = {0, 0, 0, 0};
#if defined(__clang_major__) && __clang_major__ >= 23
    i32x8 gx = {0, 0, 0, 0, 0, 0, 0, 0};
    __builtin_amdgcn_tensor_load_to_lds(g0, g1, g2, g3, gx, 0);
#else
    __builtin_amdgcn_tensor_load_to_lds(g0, g1, g2, g3, 0);
#endif
}

// ---------------------------------------------------------------------------
// Kernel 2: fused 32768x1024 fp32 WMMA GEMM + log-sum-exp.
// One block per speaker (32 rows), 8 waves, each owns a 16x16 C tile
// (rg = wave/4 row group, cw = wave%4 col stripe). 16 column iterations of
// 64 centroids; the B tile for iteration it+1 is DMA'd by the Tensor Data
// Mover into the alternate LDS buffer while iteration it runs its 64 chained
// v_wmma_f32_16x16x4_f32 (K=256). Fixed-shift (|w|+|b|) sum-exp per row.
// ---------------------------------------------------------------------------
#define LDA (D_DIM + 4)   // +4 floats row pad (matches TDM pad settings)
__global__ __launch_bounds__(256) void k_gemm_lse(
    const float* __restrict__ dvecs,
    const float* __restrict__ cent, const float* __restrict__ rcn,
    const float* __restrict__ rdn, const float* __restrict__ tlog,
    const float* __restrict__ wp, const float* __restrict__ bp,
    float* __restrict__ bsum)
{
    __shared__ float lA[32][LDA];        // ~33 KB, loaded once
    __shared__ float lB[2][64][LDA];     // 2 x ~66.5 KB, TDM double buffer
    __shared__ float part[32][4];
    __shared__ float rowred[32];

    const int t = threadIdx.x;
    const int blk = blockIdx.x;          // speaker id AND diagonal column
    const int row0 = blk * 32;
    const int wave = t >> 5;
    const int lane = t & 31;
    const int rg = wave >> 2;            // row group (0/1)
    const int cw = wave & 3;             // col stripe (0..3)
    const int lm = lane & 15;
    const int hi = lane >> 4;            // lane half -> K offset 0 or 2
    const float w = *wp, b = *bp;
    const float shift = fabsf(w) + fabsf(b);   // >= max logit (cos <= 1)

    // Stage A tile (32 rows x 256) once: 8 threads/row, 8 float4 each.
    {
        const int r = t >> 3, c = t & 7;
        const float4* s4 = (const float4*)(dvecs + (size_t)(row0 + r) * D_DIM);
        float4* d4 = (float4*)&lA[r][0];
#pragma unroll
        for (int i = 0; i < 8; ++i) d4[c + 8 * i] = s4[c + 8 * i];
    }

    // Kick off the first B tile DMA (wave 0 owns the TDM + its TENSORcnt).
    if (wave == 0) {
        tdm_load_b_tile((unsigned)(uintptr_t)&lB[0][0][0], cent);
        __builtin_amdgcn_s_wait_tensorcnt(0);
    }
    __syncthreads();

    // Per-lane row constants. C layout: VGPR j <-> tile row (hi*8 + j).
    const int myrow = rg * 16 + hi * 8;
    float rdnv[8], tlv[8];
#pragma unroll
    for (int j = 0; j < 8; ++j) {
        const int r = row0 + myrow + j;
        rdnv[j] = rdn[r];
        tlv[j]  = tlog[r];
    }
    float ssum[8];
#pragma unroll
    for (int j = 0; j < 8; ++j) ssum[j] = 0.f;

    for (int it = 0; it < N_SPK / 64; ++it) {
        const int col0 = it * 64;
        const int buf = it & 1;

        // Prefetch next B tile into the other buffer while we compute.
        if (wave == 0 && it + 1 < N_SPK / 64) {
            tdm_load_b_tile((unsigned)(uintptr_t)&lB[buf ^ 1][0][0],
                            cent + (size_t)(col0 + 64) * D_DIM);
        }

        // 16x16 fp32 WMMA tile, K = 256 in steps of 4.
        // A frag: lane -> A[lm][k+hi*2..+1]; B frag: lane -> cent[col][k+hi*2..+1]
        v8f acc = {};
        const float* arow = &lA[rg * 16 + lm][0];
        const float* brow = &lB[buf][cw * 16 + lm][0];
        const int koff = hi * 2;
#pragma unroll
        for (int k = 0; k < D_DIM; k += 4) {
            v2f a  = *(const v2f*)(arow + k + koff);
            v2f bb = *(const v2f*)(brow + k + koff);
            acc = __builtin_amdgcn_wmma_f32_16x16x4_f32(
                false, a, false, bb, (short)0, acc, false, false);
        }

        // Fused epilogue: cosine scale, clamp, diagonal swap, exp, 16-lane sum.
        const int kcol = col0 + cw * 16 + lm;
        const float rc = rcn[kcol];
#pragma unroll
        for (int j = 0; j < 8; ++j) {
            float cosv = acc[j] * rdnv[j] * rc;
            cosv = fmaxf(cosv, EPS_C);
            float l = fmaf(w, cosv, b);
            if (kcol == blk) l = tlv[j];        // leave-one-out diagonal
            float e = __expf(l - shift);
            e += __shfl_xor(e, 1, 32);
            e += __shfl_xor(e, 2, 32);
            e += __shfl_xor(e, 4, 32);
            e += __shfl_xor(e, 8, 32);          // sum over the 16 cols (half-wave)
            ssum[j] += e;
        }

        // Next tile must be resident before anyone reads it.
        if (wave == 0) __builtin_amdgcn_s_wait_tensorcnt(0);
        __syncthreads();
    }

    // Per-row partials: lanes 0 and 16 hold each half's (uniform) sums.
    if (lm == 0) {
#pragma unroll
        for (int j = 0; j < 8; ++j) part[myrow + j][cw] = ssum[j];
    }
    __syncthreads();
    if (t < 32) {
        const float tot = part[t][0] + part[t][1] + part[t][2] + part[t][3];
        const int r = row0 + t;
        rowred[t] = __logf(tot) + shift - tlog[r];   // logsumexp - target logit
    }
    __syncthreads();
    if (t == 0) {
        float tot = 0.f;
#pragma unroll
        for (int i = 0; i < 32; ++i) tot += rowred[i];
        bsum[blk] = tot;
    }
}

// ---------------------------------------------------------------------------
// Kernel 3: reduce the 1024 block partials -> mean loss.
// ---------------------------------------------------------------------------
__global__ __launch_bounds__(256) void k_final(
    const float* __restrict__ bsum, float* __restrict__ out)
{
    __shared__ float red[8];
    const int t = threadIdx.x;
    float s = 0.f;
    for (int i = t; i < R_ROWS / 32; i += 256) s += bsum[i];
#pragma unroll
    for (int off = 16; off > 0; off >>= 1) s += __shfl_xor(s, off, 32);
    if ((t & 31) == 0) red[t >> 5] = s;
    __syncthreads();
    if (t == 0) {
        float tot = 0.f;
#pragma unroll
        for (int i = 0; i < 8; ++i) tot += red[i];
        out[0] = tot / (float)R_ROWS;
    }
}

// ---------------------------------------------------------------------------
// Workspace layout (floats): cent N*D | rcn N | rdn R | tlog R | bsum R/32
// total ~1.26 MB.
// ---------------------------------------------------------------------------
extern "C" void kernel_launch(void* const* d_in, const int* in_sizes, int n_in,
                              void* d_out, int out_size, void* d_ws, size_t ws_size,
                              hipStream_t stream) {
    const float* dvecs = (const float*)d_in[0];
    const float* wp    = (const float*)d_in[1];
    const float* bp    = (const float*)d_in[2];
    // d_in[3] (label) is unused by the math.

    float* ws   = (float*)d_ws;
    float* cent = ws;
    float* rcn  = cent + (size_t)N_SPK * D_DIM;
    float* rdn  = rcn + N_SPK;
    float* tlog = rdn + R_ROWS;
    float* bsum = tlog + R_ROWS;

    k_stats<<<N_SPK, 256, 0, stream>>>(dvecs, wp, bp, cent, rcn, rdn, tlog);
    k_gemm_lse<<<R_ROWS / 32, 256, 0, stream>>>(dvecs, cent, rcn, rdn, tlog,
                                                wp, bp, bsum);
    k_final<<<1, 256, 0, stream>>>(bsum, (float*)d_out);
}